// CentroidInstanceLoss_74646531605095
// MI455X (gfx1250) — compile-verified
//
#include <hip/hip_runtime.h>
#include <math.h>

// ---------------------------------------------------------------------------
// CentroidInstanceLoss for MI455X (gfx1250, wave32)
//   N=262144 pts, D=256, S=8 subbatches, L=64 labels, 512 segments.
// Memory-bound: 2 passes over 268MB of fp32 embeddings (~23us @ 23.3TB/s).
// WMMA f32_16x16x4 (exact fp32) is used as a 16-point x 4-dim reduction
// engine for the per-point L1 distance in the pull kernel.
// ---------------------------------------------------------------------------

#define DIM   256
#define NSUB  8
#define NLBL  64
#define NSEG  (NSUB * NLBL)        // 512
#define DV    0.5f
#define DD    1.5f

// workspace layout (floats)
constexpr int OFF_SUMS   = 0;                        // [512][256]
constexpr int OFF_MUS    = NSEG * DIM;               // [512][256]
constexpr int OFF_COUNTS = 2 * NSEG * DIM;           // [512]
constexpr int OFF_M      = OFF_COUNTS + NSEG;        // [8]
constexpr int OFF_LPULL  = OFF_M + NSUB;             // [8]
constexpr int OFF_LPUSH  = OFF_LPULL + NSUB;         // [8]
constexpr int OFF_INVN   = OFF_LPUSH + NSUB;         // [N]

typedef __attribute__((ext_vector_type(2))) float v2f;
typedef __attribute__((ext_vector_type(8))) float v8f;

// ---- K0: zero the accumulator region -------------------------------------
__global__ void k_zero(float* ws, int n) {
    int i = blockIdx.x * blockDim.x + threadIdx.x;
    if (i < n) ws[i] = 0.0f;
}

// ---- K1: normalize + scatter-accumulate segment sums/counts ---------------
// one wave32 per point; lane handles dims {lane + 32*j}
__global__ void k_norm_scatter(const float* __restrict__ outputs,
                               const int* __restrict__ labels,
                               const int* __restrict__ sbidx,
                               float* ws, int n) {
    int wave = blockIdx.x * (blockDim.x >> 5) + (threadIdx.x >> 5);
    if (wave >= n) return;
    int lane = threadIdx.x & 31;

    const float* row = outputs + (size_t)wave * DIM;
    float o[8];
    float ss = 0.0f;
#pragma unroll
    for (int j = 0; j < 8; ++j) {
        o[j] = row[lane + 32 * j];           // coalesced 128B per step
        ss += o[j] * o[j];
    }
#pragma unroll
    for (int off = 16; off > 0; off >>= 1)   // wave32 butterfly reduce
        ss += __shfl_xor(ss, off, 32);
    float invn = 1.0f / (sqrtf(ss) + 1e-8f);

    int seg = sbidx[wave] * NLBL + labels[wave];
    if (lane == 0) {
        ws[OFF_INVN + wave] = invn;
        atomicAdd(&ws[OFF_COUNTS + seg], 1.0f);
    }
    float* srow = ws + OFF_SUMS + (size_t)seg * DIM;
#pragma unroll
    for (int j = 0; j < 8; ++j)              // L2-resident coalesced atomics
        atomicAdd(&srow[lane + 32 * j], o[j] * invn);
}

// ---- K2: centroids + M per subbatch ---------------------------------------
__global__ void k_centroid(float* ws) {
    int seg = blockIdx.x;                    // 512 blocks x 256 threads
    int t   = threadIdx.x;
    float c = ws[OFF_COUNTS + seg];
    ws[OFF_MUS + (size_t)seg * DIM + t] =
        ws[OFF_SUMS + (size_t)seg * DIM + t] / fmaxf(c, 1.0f);
    if (t == 0 && c > 0.0f)
        atomicAdd(&ws[OFF_M + seg / NLBL], 1.0f);
}

// ---- K3: pull term via WMMA f32 16x16x4 row-reduction ---------------------
// One wave per 16-point tile. A-matrix layout (32-bit A 16x4):
//   lanes 0-15 : M=lane,    VGPR0->K=0, VGPR1->K=1
//   lanes 16-31: M=lane-16, VGPR0->K=2, VGPR1->K=3
// so lane loads dims {4k + 2*hi, 4k + 2*hi + 1}. B = ones => C col N holds
// per-row sums; lane 0 has rows 0-7 in c[0..7], lane 16 has rows 8-15.
__global__ void k_pull(const float* __restrict__ outputs,
                       const int* __restrict__ labels,
                       const int* __restrict__ sbidx,
                       float* ws, int ntiles) {
    int wave = blockIdx.x * (blockDim.x >> 5) + (threadIdx.x >> 5);
    if (wave >= ntiles) return;              // wave-uniform: EXEC stays full
    int lane = threadIdx.x & 31;
    int p16  = lane & 15;
    int hi   = lane >> 4;
    int p    = wave * 16 + p16;

    int   sb   = sbidx[p];
    int   seg  = sb * NLBL + labels[p];
    float invn = ws[OFF_INVN + p];
    float nm   = ws[OFF_COUNTS + seg];       // Nm (>=1: own segment)
    float msb  = ws[OFF_M + sb];

    const float* orow = outputs + (size_t)p * DIM + 2 * hi;
    const float* mrow = ws + OFF_MUS + (size_t)seg * DIM + 2 * hi;

    v8f c = {};
    v2f bone; bone.x = 1.0f; bone.y = 1.0f;  // B = all-ones (exact fp32 x1.0)
#pragma unroll 4
    for (int k = 0; k < 64; ++k) {
        float o0 = orow[4 * k + 0];
        float o1 = orow[4 * k + 1];
        float m0 = mrow[4 * k + 0];
        float m1 = mrow[4 * k + 1];
        v2f a;
        a.x = fabsf(m0 - o0 * invn);
        a.y = fabsf(m1 - o1 * invn);
        // D = A x ones + C : accumulates per-point L1 distance in fp32
        c = __builtin_amdgcn_wmma_f32_16x16x4_f32(
                false, a, false, bone, (short)0, c, false, false);
    }

    // Finalize: lanes 0 / 16 own column N=0 => rows 0-7 / 8-15.
#pragma unroll
    for (int j = 0; j < 8; ++j) {
        int   src  = j + (hi << 3);          // lane0 pulls pt j, lane16 pt 8+j
        float nmj  = __shfl(nm,  src, 32);
        float msbj = __shfl(msb, src, 32);
        int   sbj  = __shfl(sb,  src, 32);
        if (p16 == 0) {                      // only lanes 0 and 16 commit
            float t = fmaxf(c[j] - DV, 0.0f);
            atomicAdd(&ws[OFF_LPULL + sbj], t * t / (msbj * nmj));
        }
    }
}

// ---- K4: push term (pairwise centroid L1 within subbatch, LDS-staged) -----
__global__ void k_push(float* ws) {
    int sb = blockIdx.x;                     // 8 blocks x 256 threads
    int t  = threadIdx.x;
    __shared__ float tile[NLBL][64 + 1];     // 64 labels x 64-dim chunk
    __shared__ float pres[NLBL];
    __shared__ float red[256];

    if (t < NLBL)
        pres[t] = (ws[OFF_COUNTS + sb * NLBL + t] > 0.0f) ? 1.0f : 0.0f;

    float acc[16];
#pragma unroll
    for (int i = 0; i < 16; ++i) acc[i] = 0.0f;

    const float* mu = ws + OFF_MUS + (size_t)sb * NLBL * DIM;
    for (int ch = 0; ch < 4; ++ch) {
        __syncthreads();
        for (int i = t; i < NLBL * 64; i += 256) {
            int a = i >> 6, d = i & 63;
            tile[a][d] = mu[(size_t)a * DIM + ch * 64 + d];
        }
        __syncthreads();
#pragma unroll
        for (int i = 0; i < 16; ++i) {
            int pair = t + i * 256;          // 4096 pairs
            int a = pair >> 6, b = pair & 63;
            float s = 0.0f;
            for (int d = 0; d < 64; ++d)
                s += fabsf(tile[a][d] - tile[b][d]);
            acc[i] += s;
        }
    }
    __syncthreads();

    float local = 0.0f;
#pragma unroll
    for (int i = 0; i < 16; ++i) {
        int pair = t + i * 256;
        int a = pair >> 6, b = pair & 63;
        if (a != b && pres[a] > 0.0f && pres[b] > 0.0f) {
            float r = fmaxf(2.0f * DD - acc[i], 0.0f);
            local += r * r;
        }
    }
    red[t] = local;
    __syncthreads();
    for (int s = 128; s > 0; s >>= 1) {
        if (t < s) red[t] += red[t + s];
        __syncthreads();
    }
    if (t == 0) {
        float M = ws[OFF_M + sb];
        ws[OFF_LPUSH + sb] = red[0] / fmaxf(M * (M - 1.0f), 1.0f);
    }
}

// ---- K5: combine -----------------------------------------------------------
__global__ void k_final(const float* __restrict__ ws, float* out, float invN) {
    float s = 0.0f;
    for (int i = 0; i < NSUB; ++i) {
        float M = ws[OFF_M + i];
        if (M > 1.0f) s += ws[OFF_LPULL + i] + ws[OFF_LPUSH + i];
    }
    out[0] = s * invN;
}

// ---------------------------------------------------------------------------
extern "C" void kernel_launch(void* const* d_in, const int* in_sizes, int n_in,
                              void* d_out, int out_size, void* d_ws, size_t ws_size,
                              hipStream_t stream) {
    const float* outputs = (const float*)d_in[0];   // [N, 256] f32
    const int*   labels  = (const int*)d_in[1];     // [N] i32
    const int*   sbidx   = (const int*)d_in[2];     // [N] i32
    float*       out     = (float*)d_out;           // scalar f32
    float*       ws      = (float*)d_ws;
    const int n = in_sizes[1];                      // N (points)

    // zero sums/counts/M/Lpull/Lpush accumulators
    k_zero<<<(OFF_INVN + 255) / 256, 256, 0, stream>>>(ws, OFF_INVN);

    // one wave per point, 8 waves/block
    k_norm_scatter<<<(n + 7) / 8, 256, 0, stream>>>(outputs, labels, sbidx, ws, n);

    k_centroid<<<NSEG, DIM, 0, stream>>>(ws);

    int ntiles = n / 16;                            // N divisible by 16
    k_pull<<<(ntiles + 7) / 8, 256, 0, stream>>>(outputs, labels, sbidx, ws, ntiles);

    k_push<<<NSUB, 256, 0, stream>>>(ws);

    k_final<<<1, 1, 0, stream>>>(ws, out, 1.0f / (float)n);
}